// Target_v1_63702954934446
// MI455X (gfx1250) — compile-verified
//
#include <hip/hip_runtime.h>
#include <hip/hip_bf16.h>
#include <stdint.h>

// ---------------- problem constants ----------------
#define B_N      16
#define M_N      100
#define HW_TOTAL 21824            // 128^2 + 64^2 + 32^2 + 16^2 + 8^2
#define CLS_DIM  82               // 80 classes + soft + posf
#define REG_DIM  6                // 4 reg + soft + posf
#define BIGF     10000000.0f

// flat output layout (floats), concat in reference return order
#define CLS_OFF  0LL
#define REG_OFF  ((long long)B_N * HW_TOTAL * CLS_DIM)                 // 28,633,088
#define IND_OFF  (REG_OFF + (long long)B_N * HW_TOTAL * REG_DIM)      // 30,728,192
#define NB_OFF   (IND_OFF + (long long)B_N * HW_TOTAL)                // 31,077,376

typedef float vfloat4 __attribute__((ext_vector_type(4)));

// ---------------- kernel A: zero-fill of cls region (RT stores: stays L2-resident
// on MI455X's 192MB L2 so the later scatter writes merge without HBM refetch) ----
__global__ __launch_bounds__(256) void zerofill_kernel(vfloat4* __restrict__ p, long long n4) {
    long long i   = (long long)blockIdx.x * blockDim.x + threadIdx.x;
    long long str = (long long)gridDim.x * blockDim.x;
    vfloat4 z = (vfloat4)(0.0f);
    for (; i < n4; i += str)
        p[i] = z;                       // global_store_b128
}

// ---------------- kernel B: per-cell target assignment ----------------
__global__ __launch_bounds__(256) void fcos_assign_kernel(const float* __restrict__ gt,
                                                          float* __restrict__ out) {
    const int batch = blockIdx.y;
    const int tid   = threadIdx.x;

    __shared__ __align__(16) float raw[M_N * 5];   // x1,y1,x2,y2,label per box (async b128 dest)
    __shared__ float4 bnd[5 * M_N];                // per (level,box): px1,py1,px2,py2
    __shared__ float  s_area[M_N];
    __shared__ float  s_x1[M_N], s_y1[M_N], s_x2[M_N], s_y2[M_N], s_lab[M_N];

    // ---- CDNA5 async copy: 500 floats (2000 B = 125 x b128, 16B aligned) global -> LDS ----
    if (tid < 125) {
        const float* g = gt + (long long)batch * (M_N * 5) + tid * 4;
        unsigned loff = (unsigned)(uintptr_t)(&raw[tid * 4]);   // low 32 bits = LDS offset
        asm volatile("global_load_async_to_lds_b128 %0, %1, off"
                     :: "v"(loff), "v"(g) : "memory");
    }
    asm volatile("s_wait_asynccnt 0" ::: "memory");
    __syncthreads();

    // ---- per-box raw fields ----
    if (tid < M_N) {
        float x1 = raw[tid * 5 + 0], y1 = raw[tid * 5 + 1];
        float x2 = raw[tid * 5 + 2], y2 = raw[tid * 5 + 3];
        s_x1[tid] = x1; s_y1[tid] = y1; s_x2[tid] = x2; s_y2[tid] = y2;
        s_lab[tid] = raw[tid * 5 + 4];
        s_area[tid] = (x2 - x1) * (y2 - y1);
    }
    // ---- per (level,box) shrunk center regions ----
    for (int idx = tid; idx < 5 * M_N; idx += 256) {
        int lv = idx / M_N, m = idx - lv * M_N;
        float stride = (float)(8 << lv);
        float fwf    = (float)(128 >> lv);
        float x1 = raw[m * 5 + 0], y1 = raw[m * 5 + 1];
        float x2 = raw[m * 5 + 2], y2 = raw[m * 5 + 3];
        bool valid = (fabsf(x1) + fabsf(y1) + fabsf(x2) + fabsf(y2)) > 0.0f;
        float inv = 1.0f / stride;
        float bx1 = x1 * inv, by1 = y1 * inv, bx2 = x2 * inv, by2 = y2 * inv;
        float cx = (bx1 + bx2) * 0.5f, cy = (by1 + by2) * 0.5f;
        float hw = (bx2 - bx1) * 0.5f * 0.2f, hh = (by2 - by1) * 0.5f * 0.2f;
        float px1 = fmaxf(floorf(cx - hw), 0.0f);
        float py1 = fmaxf(floorf(cy - hh), 0.0f);
        float px2 = fminf(ceilf(cx + hw), fwf);
        float py2 = fminf(ceilf(cy + hh), fwf);
        if (!valid) { px1 = 3.0e9f; px2 = -3.0e9f; }
        bnd[idx] = make_float4(px1, py1, px2, py2);
    }
    __syncthreads();

    // ---- per-batch valid-box count (reference: num_boxes), done once per batch ----
    if (blockIdx.x == 0 && tid == 0) {
        int cnt = 0;
        for (int m = 0; m < M_N; ++m) {
            float s = fabsf(raw[m * 5 + 0]) + fabsf(raw[m * 5 + 1]) +
                      fabsf(raw[m * 5 + 2]) + fabsf(raw[m * 5 + 3]);
            cnt += (s > 0.0f) ? 1 : 0;
        }
        out[NB_OFF + batch] = (float)cnt;
    }

    const int cell = blockIdx.x * 256 + tid;
    if (cell >= HW_TOTAL) return;

    // ---- which pyramid level ----
    int lv, base;
    if      (cell < 16384) { lv = 0; base = 0;     }
    else if (cell < 20480) { lv = 1; base = 16384; }
    else if (cell < 21504) { lv = 2; base = 20480; }
    else if (cell < 21760) { lv = 3; base = 21504; }
    else                   { lv = 4; base = 21760; }
    const int shift = 7 - lv;                   // log2(fw)
    const int local = cell - base;
    const int yi = local >> shift;
    const int xi = local & ((1 << shift) - 1);
    const float xf = (float)xi, yf = (float)yi;

    // ---- argmin over boxes (strict <: first-min tie-break, matches jnp.argmin) ----
    float best = BIGF;
    int   win  = 0;
    const float4* bl = &bnd[lv * M_N];
    #pragma unroll 4
    for (int m = 0; m < M_N; ++m) {
        float4 b = bl[m];                       // ds_load_b128, wave-broadcast
        bool inside = (xf >= b.x) & (xf < b.z) & (yf >= b.y) & (yf < b.w);
        float a = inside ? s_area[m] : BIGF;
        if (a < best) { best = a; win = m; }
    }
    const bool  pos  = best < BIGF;
    const float posf = pos ? 1.0f : 0.0f;

    const float stridef = (float)(8 << lv);
    const float sx = (xf + 0.5f) * stridef;
    const float sy = (yf + 0.5f) * stridef;
    const float l   = sx - s_x1[win];
    const float t   = sy - s_y1[win];
    const float r   = s_x2[win] - sx;
    const float btm = s_y2[win] - sy;
    const float inv4s = 1.0f / (4.0f * stridef);
    const float eps = 1e-6f;
    float rx = fminf(l, r)   / fmaxf(fmaxf(l, r),   eps);
    float ry = fminf(t, btm) / fmaxf(fmaxf(t, btm), eps);
    rx = fminf(fmaxf(rx, 0.0f), 1.0f);
    ry = fminf(fmaxf(ry, 0.0f), 1.0f);
    const float cent = sqrtf(rx * ry);
    const float soft = pos ? cent : 1.0f;

    const long long gcell = (long long)batch * HW_TOTAL + cell;

    // cls row: region pre-zeroed; write one-hot (if pos) + [soft, posf]
    float* clsrow = out + CLS_OFF + gcell * CLS_DIM;
    if (pos) clsrow[(int)s_lab[win]] = 1.0f;
    *(float2*)(clsrow + 80) = make_float2(soft, posf);

    // reg row: [l,t,r,b]*posf/(4*stride), soft, posf
    float* regrow = out + REG_OFF + gcell * REG_DIM;
    ((float2*)regrow)[0] = make_float2(l   * inv4s * posf, t   * inv4s * posf);
    ((float2*)regrow)[1] = make_float2(r   * inv4s * posf, btm * inv4s * posf);
    ((float2*)regrow)[2] = make_float2(soft, posf);

    // index (exact small ints in float)
    out[IND_OFF + gcell] = pos ? (float)win : -1.0f;
}

// ---------------- launch ----------------
extern "C" void kernel_launch(void* const* d_in, const int* in_sizes, int n_in,
                              void* d_out, int out_size, void* d_ws, size_t ws_size,
                              hipStream_t stream) {
    (void)in_sizes; (void)n_in; (void)out_size; (void)d_ws; (void)ws_size;
    const float* gt  = (const float*)d_in[0];     // [16,100,5] float32
    float*       out = (float*)d_out;

    // 1) zero cls region (16*21824*82 floats, divisible by 4)
    const long long n4 = REG_OFF / 4;
    zerofill_kernel<<<8192, 256, 0, stream>>>((vfloat4*)out, n4);

    // 2) per-cell assignment (+ per-batch counts): grid = (ceil(21824/256)=86, B=16)
    dim3 grid((HW_TOTAL + 255) / 256, B_N);
    fcos_assign_kernel<<<grid, 256, 0, stream>>>(gt, out);
}